// GraphSAGEModel_39676907888759
// MI455X (gfx1250) — compile-verified
//
#include <hip/hip_runtime.h>
#include <hip/hip_bf16.h>

#define H 128
#define NGRAPH 512
#define NCLS 6
#define LN_EPS 1e-5f

typedef float v2f __attribute__((ext_vector_type(2)));
typedef float v8f __attribute__((ext_vector_type(8)));

// ---------------------------------------------------------------------------
// f32 WMMA helper: D = A(16x4) * B(4x16) + C, exact fp32 accumulation.
// 8-arg form: (neg_a, A, neg_b, B, c_mod, C, reuse_a, reuse_b)
// ---------------------------------------------------------------------------
__device__ __forceinline__ v8f wmma_f32(v2f a, v2f b, v8f c) {
    return __builtin_amdgcn_wmma_f32_16x16x4_f32(false, a, false, b, (short)0, c,
                                                 false, false);
}

__device__ __forceinline__ float elu_f(float x) {
    return x > 0.f ? x : (expf(x) - 1.f);
}

// monotonic float<->uint encoding for atomicMax on floats
__device__ __forceinline__ unsigned enc_f(float f) {
    unsigned u = __float_as_uint(f);
    return (u & 0x80000000u) ? ~u : (u | 0x80000000u);
}
__device__ __forceinline__ float dec_f(unsigned e) {
    return __uint_as_float((e & 0x80000000u) ? (e & 0x7fffffffu) : ~e);
}

// ---------------------------------------------------------------------------
// generic zero
// ---------------------------------------------------------------------------
__global__ void zero_kernel(unsigned* __restrict__ p, int n) {
    int i = blockIdx.x * 256 + threadIdx.x;
    if (i < n) p[i] = 0u;
}

// ---------------------------------------------------------------------------
// CSR build: degree count -> exclusive scan -> fill
// ---------------------------------------------------------------------------
__global__ void deg_count_kernel(const int* __restrict__ dst, int* __restrict__ cnt, int e) {
    int i = blockIdx.x * 256 + threadIdx.x;
    if (i < e) atomicAdd(&cnt[dst[i]], 1);
}

__global__ void scan_kernel(const int* __restrict__ cnt, int* __restrict__ rowptr, int n) {
    __shared__ int buf[1024];
    __shared__ int carry;
    int tid = threadIdx.x;
    if (tid == 0) carry = 0;
    __syncthreads();
    for (int base = 0; base < n; base += 1024) {
        int i = base + tid;
        int v = (i < n) ? cnt[i] : 0;
        buf[tid] = v;
        __syncthreads();
        for (int off = 1; off < 1024; off <<= 1) {   // Hillis-Steele inclusive scan
            int t = (tid >= off) ? buf[tid - off] : 0;
            __syncthreads();
            buf[tid] += t;
            __syncthreads();
        }
        int incl = buf[tid];
        int c = carry;
        if (i < n) rowptr[i] = c + incl - v;          // exclusive prefix
        __syncthreads();
        if (tid == 0) carry = c + buf[1023];
        __syncthreads();
    }
    if (tid == 0) rowptr[n] = carry;
}

__global__ void csr_fill_kernel(const int* __restrict__ src, const int* __restrict__ dst,
                                const int* __restrict__ rowptr, int* __restrict__ fill,
                                int* __restrict__ colidx, int e) {
    int i = blockIdx.x * 256 + threadIdx.x;
    if (i < e) {
        int d = dst[i];
        int pos = rowptr[d] + atomicAdd(&fill[d], 1);
        colidx[pos] = src[i];
    }
}

// ---------------------------------------------------------------------------
// Projection: h = relu(x @ W[64,128] + b). One block = 16 rows, 8 waves;
// wave w owns 16-column tile w. K=64 via v_wmma_f32_16x16x4_f32.
// ---------------------------------------------------------------------------
__global__ void proj_kernel(const float* __restrict__ x, const float* __restrict__ w,
                            const float* __restrict__ bias, float* __restrict__ h) {
    __shared__ float As[16 * 64];
    int tid = threadIdx.x;
    int rowbase = blockIdx.x * 16;
    ((float4*)As)[tid] = ((const float4*)(x + (size_t)rowbase * 64))[tid];
    __syncthreads();

    int lane = tid & 31, wid = tid >> 5;
    int hf = lane >> 4, m = lane & 15;
    int col = wid * 16 + m;
    v8f acc = {};
#pragma unroll
    for (int kk = 0; kk < 64; kk += 4) {
        int k0 = kk + 2 * hf;
        v2f a = *(v2f*)(&As[m * 64 + k0]);           // A: lanes0-15 K={kk,kk+1}, lanes16-31 K={kk+2,kk+3}
        v2f b;                                       // B mirrored over N-lanes
        b.x = w[(size_t)k0 * H + col];
        b.y = w[(size_t)(k0 + 1) * H + col];
        acc = wmma_f32(a, b, acc);
    }
    float bv = bias[col];
#pragma unroll
    for (int r = 0; r < 8; ++r) {                    // C/D: vgpr r -> row r + 8*half
        int row = rowbase + r + 8 * hf;
        float v = acc[r] + bv;
        h[(size_t)row * H + col] = v > 0.f ? v : 0.f;
    }
}

// ---------------------------------------------------------------------------
// Mean aggregation: one wave per destination node, gather CSR neighbors.
// h fits in 192MB L2 (51.2MB), so these are L2-resident streaming gathers.
// ---------------------------------------------------------------------------
__global__ void aggr_kernel(const float* __restrict__ h, const int* __restrict__ rowptr,
                            const int* __restrict__ colidx, float* __restrict__ out, int n) {
    int wid = threadIdx.x >> 5, lane = threadIdx.x & 31;
    int node = blockIdx.x * 8 + wid;
    if (node >= n) return;
    int s = rowptr[node], e = rowptr[node + 1];
    float4 acc = {0.f, 0.f, 0.f, 0.f};
    for (int i = s; i < e; ++i) {
        int src = colidx[i];
        float4 v = ((const float4*)(h + (size_t)src * H))[lane];
        acc.x += v.x; acc.y += v.y; acc.z += v.z; acc.w += v.w;
    }
    float inv = 1.f / (float)max(e - s, 1);
    acc.x *= inv; acc.y *= inv; acc.z *= inv; acc.w *= inv;
    ((float4*)(out + (size_t)node * H))[lane] = acc;
}

// ---------------------------------------------------------------------------
// Fused SAGE layer: out = elu(LN(aggr@Wl + bl + h@Wr)) + h.
// 16-row tile in LDS; 8 waves x 16-col WMMA tiles (both GEMMs accumulate into
// the same v8f); LN/ELU/residual done via wave shuffles; hout may alias aggr
// (each block stages its own rows to LDS before overwriting them).
// ---------------------------------------------------------------------------
__global__ void sage_layer_kernel(const float* hin, const float* aggr,
                                  const float* __restrict__ wl, const float* __restrict__ bl,
                                  const float* __restrict__ wr,
                                  const float* __restrict__ lng, const float* __restrict__ lnb,
                                  float* hout) {
    __shared__ float Aag[16 * H];
    __shared__ float Ah[16 * H];
    __shared__ float Out[16 * H];
    int tid = threadIdx.x;
    int rowbase = blockIdx.x * 16;
    {
        const float4* ga = (const float4*)(aggr + (size_t)rowbase * H);
        const float4* gh = (const float4*)(hin + (size_t)rowbase * H);
        ((float4*)Aag)[tid]       = ga[tid];
        ((float4*)Aag)[tid + 256] = ga[tid + 256];
        ((float4*)Ah)[tid]        = gh[tid];
        ((float4*)Ah)[tid + 256]  = gh[tid + 256];
    }
    __syncthreads();

    int lane = tid & 31, wid = tid >> 5;
    int hf = lane >> 4, m = lane & 15;
    int col = wid * 16 + m;
    v8f acc = {};
#pragma unroll
    for (int kk = 0; kk < H; kk += 4) {
        int k0 = kk + 2 * hf;
        v2f aa = *(v2f*)(&Aag[m * H + k0]);
        v2f bL; bL.x = wl[(size_t)k0 * H + col]; bL.y = wl[(size_t)(k0 + 1) * H + col];
        acc = wmma_f32(aa, bL, acc);
        v2f ah = *(v2f*)(&Ah[m * H + k0]);
        v2f bR; bR.x = wr[(size_t)k0 * H + col]; bR.y = wr[(size_t)(k0 + 1) * H + col];
        acc = wmma_f32(ah, bR, acc);
    }
    float bv = bl[col];
#pragma unroll
    for (int r = 0; r < 8; ++r)
        Out[(r + 8 * hf) * H + col] = acc[r] + bv;
    __syncthreads();

    // LayerNorm + ELU + residual: wave `wid` handles rows 2*wid, 2*wid+1
#pragma unroll
    for (int rr = 0; rr < 2; ++rr) {
        int row = wid * 2 + rr;
        float4 v = ((float4*)&Out[row * H])[lane];
        float s = v.x + v.y + v.z + v.w;
        for (int off = 16; off > 0; off >>= 1) s += __shfl_xor(s, off);
        float mu = s * (1.f / 128.f);
        float dx = v.x - mu, dy = v.y - mu, dz = v.z - mu, dw = v.w - mu;
        float q = dx * dx + dy * dy + dz * dz + dw * dw;
        for (int off = 16; off > 0; off >>= 1) q += __shfl_xor(q, off);
        float rstd = rsqrtf(q * (1.f / 128.f) + LN_EPS);
        float4 gv = ((const float4*)lng)[lane];
        float4 bb = ((const float4*)lnb)[lane];
        float4 r4 = ((float4*)&Ah[row * H])[lane];
        float4 o;
        o.x = elu_f(dx * rstd * gv.x + bb.x) + r4.x;
        o.y = elu_f(dy * rstd * gv.y + bb.y) + r4.y;
        o.z = elu_f(dz * rstd * gv.z + bb.z) + r4.z;
        o.w = elu_f(dw * rstd * gv.w + bb.w) + r4.w;
        ((float4*)(hout + (size_t)(rowbase + row) * H))[lane] = o;
    }
}

// ---------------------------------------------------------------------------
// Gate MLP + segment max: one wave per node; lane owns hidden cols {lane, lane+32}
// ---------------------------------------------------------------------------
__global__ void gate_kernel(const float* __restrict__ h, const float* __restrict__ w1,
                            const float* __restrict__ b1, const float* __restrict__ w2,
                            const float* __restrict__ b2, const int* __restrict__ batch,
                            float* __restrict__ gate, unsigned* __restrict__ gmax, int n) {
    int wid = threadIdx.x >> 5, lane = threadIdx.x & 31;
    int node = blockIdx.x * 8 + wid;
    if (node >= n) return;
    const float* hr = h + (size_t)node * H;
    float a0 = 0.f, a1 = 0.f;
    for (int k = 0; k < H; ++k) {
        float hv = hr[k];                  // broadcast load (L0/L2 cached)
        a0 += hv * w1[k * 64 + lane];
        a1 += hv * w1[k * 64 + lane + 32];
    }
    a0 = fmaxf(a0 + b1[lane], 0.f);
    a1 = fmaxf(a1 + b1[lane + 32], 0.f);
    float gv = a0 * w2[lane] + a1 * w2[lane + 32];
    for (int off = 16; off > 0; off >>= 1) gv += __shfl_xor(gv, off);
    gv += b2[0];
    if (lane == 0) {
        gate[node] = gv;
        atomicMax(&gmax[batch[node]], enc_f(gv));
    }
}

__global__ void softmax_den_kernel(const int* __restrict__ batch, const unsigned* __restrict__ gmax,
                                   float* __restrict__ gate, float* __restrict__ denom, int n) {
    int i = blockIdx.x * 256 + threadIdx.x;
    if (i < n) {
        int b = batch[i];
        float e = expf(gate[i] - dec_f(gmax[b]));
        gate[i] = e;                        // in-place: eg
        atomicAdd(&denom[b], e);
    }
}

__global__ void pool_kernel(const float* __restrict__ h, const float* __restrict__ eg,
                            const float* __restrict__ denom, const int* __restrict__ batch,
                            float* pooled, int n) {
    int wid = threadIdx.x >> 5, lane = threadIdx.x & 31;
    int node = blockIdx.x * 8 + wid;
    if (node >= n) return;
    int b = batch[node];
    float alpha = eg[node] / denom[b];
    float4 v = ((const float4*)(h + (size_t)node * H))[lane];
    float* pb = pooled + (size_t)b * H + lane * 4;
    atomicAdd(pb + 0, v.x * alpha);
    atomicAdd(pb + 1, v.y * alpha);
    atomicAdd(pb + 2, v.z * alpha);
    atomicAdd(pb + 3, v.w * alpha);
}

// ---------------------------------------------------------------------------
// Classifier: one wave per graph.
// ---------------------------------------------------------------------------
__global__ void cls_kernel(const float* __restrict__ pooled, const float* __restrict__ w1,
                           const float* __restrict__ b1, const float* __restrict__ w2,
                           const float* __restrict__ b2, float* __restrict__ out, int ngr) {
    int wid = threadIdx.x >> 5, lane = threadIdx.x & 31;
    int g = blockIdx.x * 8 + wid;
    if (g >= ngr) return;
    const float* pr = pooled + (size_t)g * H;
    float a0 = 0.f, a1 = 0.f;
    for (int k = 0; k < H; ++k) {
        float pv = pr[k];
        a0 += pv * w1[k * 64 + lane];
        a1 += pv * w1[k * 64 + lane + 32];
    }
    a0 = fmaxf(a0 + b1[lane], 0.f);
    a1 = fmaxf(a1 + b1[lane + 32], 0.f);
#pragma unroll
    for (int j = 0; j < NCLS; ++j) {
        float p = a0 * w2[lane * NCLS + j] + a1 * w2[(lane + 32) * NCLS + j];
        for (int off = 16; off > 0; off >>= 1) p += __shfl_xor(p, off);
        if (lane == 0) out[g * NCLS + j] = p + b2[j];
    }
}

// ---------------------------------------------------------------------------
extern "C" void kernel_launch(void* const* d_in, const int* in_sizes, int n_in,
                              void* d_out, int out_size, void* d_ws, size_t ws_size,
                              hipStream_t stream) {
    const float* x       = (const float*)d_in[0];
    const int*   ei      = (const int*)d_in[1];
    const int*   batch   = (const int*)d_in[2];
    const float* proj_w  = (const float*)d_in[3];
    const float* proj_b  = (const float*)d_in[4];
    const float* lin_l_w = (const float*)d_in[5];
    const float* lin_l_b = (const float*)d_in[6];
    const float* lin_r_w = (const float*)d_in[7];
    const float* ln_g    = (const float*)d_in[8];
    const float* ln_b    = (const float*)d_in[9];
    const float* gate_w1 = (const float*)d_in[10];
    const float* gate_b1 = (const float*)d_in[11];
    const float* gate_w2 = (const float*)d_in[12];
    const float* gate_b2 = (const float*)d_in[13];
    const float* cls_w1  = (const float*)d_in[14];
    const float* cls_b1  = (const float*)d_in[15];
    const float* cls_w2  = (const float*)d_in[16];
    const float* cls_b2  = (const float*)d_in[17];
    float* out = (float*)d_out;

    const int N = in_sizes[0] / 64;   // 100000 (multiple of 16)
    const int E = in_sizes[1] / 2;    // 1600000
    const int* src = ei;
    const int* dst = ei + E;

    // --- workspace carving (256B aligned) ---
    char* w = (char*)d_ws;
    auto carve = [&](size_t bytes) -> void* {
        void* p = (void*)w;
        w += (bytes + 255) & ~(size_t)255;
        return p;
    };
    float*    hA     = (float*)carve((size_t)N * H * 4);
    float*    hB     = (float*)carve((size_t)N * H * 4);
    int*      rowptr = (int*)carve((size_t)(N + 1) * 4);
    int*      cnt    = (int*)carve((size_t)N * 4);       // reused as fill
    int*      colidx = (int*)carve((size_t)E * 4);
    float*    gate   = (float*)carve((size_t)N * 4);
    unsigned* gmax   = (unsigned*)carve((size_t)NGRAPH * 4);
    float*    denom  = (float*)carve((size_t)NGRAPH * 4);
    float*    pooled = (float*)carve((size_t)NGRAPH * H * 4);

    const int eBlocks = (E + 255) / 256;
    const int nBlocks = (N + 255) / 256;

    // --- CSR build (once per launch; deterministic set of work) ---
    zero_kernel<<<nBlocks, 256, 0, stream>>>((unsigned*)cnt, N);
    deg_count_kernel<<<eBlocks, 256, 0, stream>>>(dst, cnt, E);
    scan_kernel<<<1, 1024, 0, stream>>>(cnt, rowptr, N);
    zero_kernel<<<nBlocks, 256, 0, stream>>>((unsigned*)cnt, N);
    csr_fill_kernel<<<eBlocks, 256, 0, stream>>>(src, dst, rowptr, cnt, colidx, E);

    // --- projection ---
    proj_kernel<<<N / 16, 256, 0, stream>>>(x, proj_w, proj_b, hA);

    // --- 3 SAGE layers, ping-pong hA/hB (layer output overwrites aggr buffer) ---
    float* cur = hA;
    float* oth = hB;
    for (int l = 0; l < 3; ++l) {
        aggr_kernel<<<(N + 7) / 8, 256, 0, stream>>>(cur, rowptr, colidx, oth, N);
        sage_layer_kernel<<<N / 16, 256, 0, stream>>>(
            cur, oth,
            lin_l_w + (size_t)l * H * H, lin_l_b + (size_t)l * H,
            lin_r_w + (size_t)l * H * H,
            ln_g + (size_t)l * H, ln_b + (size_t)l * H,
            oth);
        float* t = cur; cur = oth; oth = t;
    }

    // --- attention pooling ---
    zero_kernel<<<(NGRAPH + 255) / 256, 256, 0, stream>>>(gmax, NGRAPH);
    zero_kernel<<<(NGRAPH + 255) / 256, 256, 0, stream>>>((unsigned*)denom, NGRAPH);
    zero_kernel<<<(NGRAPH * H + 255) / 256, 256, 0, stream>>>((unsigned*)pooled, NGRAPH * H);

    gate_kernel<<<(N + 7) / 8, 256, 0, stream>>>(cur, gate_w1, gate_b1, gate_w2, gate_b2,
                                                 batch, gate, gmax, N);
    softmax_den_kernel<<<nBlocks, 256, 0, stream>>>(batch, gmax, gate, denom, N);
    pool_kernel<<<(N + 7) / 8, 256, 0, stream>>>(cur, gate, denom, batch, pooled, N);

    // --- classifier ---
    cls_kernel<<<(NGRAPH + 7) / 8, 256, 0, stream>>>(pooled, cls_w1, cls_b1, cls_w2, cls_b2,
                                                     out, NGRAPH);
}